// SupportCalibrator_3539053052049
// MI455X (gfx1250) — compile-verified
//
#include <hip/hip_runtime.h>

// ---------------------------------------------------------------------------
// SupportCalibrator on MI455X (gfx1250), fp32 WMMA (16x16x4), wave32.
//
// Algebraic refactor: k/v projections are pulled through the einsums so the
// 64000-row key/value GEMMs (87 GFLOP, 262 MB of intermediates) become
// 128-row query-side transforms. Remaining heavy work:
//   - map_sem over base semantics: two 64000x300x300 GEMMs (WMMA)
//   - score = qk . bw^T + qs . bsc^T : 8 x (16x8000) K=512/300 (WMMA)
//   - ctx = attn @ bw : 8 x (16x512) K=8000 (WMMA, K split over 8 waves + LDS)
// Memory-bound: ~0.6 GB of HBM traffic @ 23.3 TB/s.
// Workspace requirement: 2 * 19,200,000 floats = 153.6 MB.
// ---------------------------------------------------------------------------

typedef __attribute__((ext_vector_type(2))) float v2f;
typedef __attribute__((ext_vector_type(8))) float v8f;

#define SEMD 300
#define FEATD 512
#define NBASE 8000
#define INV_TEMP 0.04419417382415922f   // 1/sqrt(512)

static __device__ __forceinline__ v8f wmma_f32(v2f a, v2f b, v8f c) {
    return __builtin_amdgcn_wmma_f32_16x16x4_f32(
        false, a, false, b, (short)0, c, false, false);
}

// Y[M x 300] = act( X[M x 300] @ W[300 x 300] + bias ),  act: leaky_relu(0.1) or identity
// grid: (M/16, 3), block 256 (8 waves); wave w covers N-tile (by*8+w)*16.
// Branchless tail handling: out-of-range lanes load a clamped column (their WMMA
// output column is garbage but is never stored).
__global__ void mapsem_kernel(const float* __restrict__ X, const float* __restrict__ W,
                              const float* __restrict__ bias, float* __restrict__ Y,
                              int act) {
    const int lane = threadIdx.x & 31;
    const int wave = threadIdx.x >> 5;
    const int m0 = blockIdx.x * 16;
    const int n0 = (blockIdx.y * 8 + wave) * 16;
    if (n0 >= SEMD) return;                 // wave-uniform
    const int ml = lane & 15;
    const int kb = (lane >> 4) * 2;         // 0 or 2
    const int nl = n0 + ml;
    const bool nv = (nl < SEMD);
    const int nls = nv ? nl : (SEMD - 1);   // clamped, always in-bounds
    const float bv = bias[nls];
    v8f acc;
#pragma unroll
    for (int r = 0; r < 8; ++r) acc[r] = bv;
    const float* __restrict__ arow = X + (size_t)(m0 + ml) * SEMD;
    for (int kk = 0; kk < SEMD; kk += 4) {
        v2f a, b;
        a[0] = arow[kk + kb];
        a[1] = arow[kk + kb + 1];
        b[0] = W[(size_t)(kk + kb) * SEMD + nls];
        b[1] = W[(size_t)(kk + kb + 1) * SEMD + nls];
        acc = wmma_f32(a, b, acc);
    }
    if (act) {
#pragma unroll
        for (int r = 0; r < 8; ++r) acc[r] = (acc[r] > 0.0f) ? acc[r] : 0.1f * acc[r];
    }
    if (nv) {
        const int mrow = m0 + 8 * (lane >> 4);
#pragma unroll
        for (int r = 0; r < 8; ++r) Y[(size_t)(mrow + r) * SEMD + nl] = acc[r];
    }
}

// Stage 1 of column means: partial sums over 500-row slices.
// part[c][b][e], grid (4, 8, 16), block 256
__global__ void means_partial_kernel(const float* __restrict__ bw, const float* __restrict__ bsc,
                                     float* __restrict__ part) {
    const int e = blockIdx.x * 256 + threadIdx.x;
    const int b = blockIdx.y;
    const int c = blockIdx.z;
    if (e >= FEATD + SEMD) return;
    const int nstart = c * 500;
    float s = 0.0f;
    if (e < FEATD) {
        const float* p = bw + ((size_t)b * NBASE + nstart) * FEATD + e;
        for (int n = 0; n < 500; ++n) s += p[(size_t)n * FEATD];
    } else {
        const float* p = bsc + ((size_t)b * NBASE + nstart) * SEMD + (e - FEATD);
        for (int n = 0; n < 500; ++n) s += p[(size_t)n * SEMD];
    }
    part[((size_t)c * 8 + b) * 812 + e] = s;
}

// Stage 2: mean[b][e] = (1/8000) * sum_c part[c][b][e]; grid (4, 8), block 256
__global__ void means_final_kernel(const float* __restrict__ part, float* __restrict__ mean) {
    const int e = blockIdx.x * 256 + threadIdx.x;
    const int b = blockIdx.y;
    if (e >= FEATD + SEMD) return;
    float s = 0.0f;
#pragma unroll
    for (int c = 0; c < 16; ++c) s += part[((size_t)c * 8 + b) * 812 + e];
    mean[b * 812 + e] = s * (1.0f / 8000.0f);
}

// gate[b][e] = sigmoid(avg . fuse_w[:,e] + fuse_b[e]) + 1   (vis for e<512, sem after)
// grid (4, 8), block 256
__global__ void gates_kernel(const float* __restrict__ mean,
                             const float* __restrict__ visw, const float* __restrict__ visb,
                             const float* __restrict__ semw, const float* __restrict__ semb,
                             float* __restrict__ gate) {
    const int e = blockIdx.x * 256 + threadIdx.x;
    const int b = blockIdx.y;
    if (e >= FEATD + SEMD) return;
    const float* avg = mean + b * 812;
    float s;
    if (e < FEATD) {
        s = visb[e];
        for (int i = 0; i < 812; ++i) s += avg[i] * visw[(size_t)i * FEATD + e];
    } else {
        const int e2 = e - FEATD;
        s = semb[e2];
        for (int i = 0; i < 812; ++i) s += avg[i] * semw[(size_t)i * SEMD + e2];
    }
    gate[b * 812 + e] = 1.0f + 1.0f / (1.0f + expf(-s));
}

// q_tot[r][d] = sf[r,:] @ w_qs[:,d] + ssc[r,:] @ w_qs_sem[:,d]   grid (2,128), block 256
__global__ void qtot_kernel(const float* __restrict__ sf, const float* __restrict__ ssc,
                            const float* __restrict__ wq, const float* __restrict__ wqs,
                            float* __restrict__ qt) {
    const int r = blockIdx.y;
    const int d = blockIdx.x * 256 + threadIdx.x;
    const float* x = sf + (size_t)r * FEATD;
    const float* y = ssc + (size_t)r * SEMD;
    float s = 0.0f;
    for (int e = 0; e < FEATD; ++e) s += x[e] * wq[(size_t)e * FEATD + d];
    for (int e = 0; e < SEMD; ++e) s += y[e] * wqs[(size_t)e * FEATD + d];
    qt[(size_t)r * FEATD + d] = s;
}

// qk[r][e]   = gv[b][e]  * (w_ks[e,:]     . qt[r,:])   e<512
// qs[r][e2]  = gs[b][e2] * (w_ks_sem[e2,:]. qt[r,:])   grid (4,128), block 256
__global__ void qk_kernel(const float* __restrict__ qt, const float* __restrict__ gate,
                          const float* __restrict__ wk, const float* __restrict__ wks,
                          float* __restrict__ qk, float* __restrict__ qs) {
    const int r = blockIdx.y;
    const int e = blockIdx.x * 256 + threadIdx.x;
    if (e >= FEATD + SEMD) return;
    const int b = r >> 4;
    const float* q = qt + (size_t)r * FEATD;
    if (e < FEATD) {
        const float* w = wk + (size_t)e * FEATD;
        float s = 0.0f;
        for (int d = 0; d < FEATD; ++d) s += w[d] * q[d];
        qk[(size_t)r * FEATD + e] = s * gate[b * 812 + e];
    } else {
        const int e2 = e - FEATD;
        const float* w = wks + (size_t)e2 * FEATD;
        float s = 0.0f;
        for (int d = 0; d < FEATD; ++d) s += w[d] * q[d];
        qs[(size_t)r * SEMD + e2] = s * gate[b * 812 + FEATD + e2];
    }
}

// score[b,w,n] = (qk[b,w,:] . bw[b,n,:] + qs[b,w,:] . bsc[b,n,:]) / sqrt(512)
// grid (63, 8), block 256 (8 waves): wave tile t = bx*8+wave, n0 = t*16, t < 500
__global__ void score_kernel(const float* __restrict__ qk, const float* __restrict__ qs,
                             const float* __restrict__ bw, const float* __restrict__ bsc,
                             float* __restrict__ score) {
    const int lane = threadIdx.x & 31;
    const int wave = threadIdx.x >> 5;
    const int b = blockIdx.y;
    const int t = blockIdx.x * 8 + wave;
    if (t >= NBASE / 16) return;            // wave-uniform
    const int n0 = t * 16;
    const int ml = lane & 15;
    const int kb = (lane >> 4) * 2;
    const int nl = n0 + ml;
    v8f acc = {};
    {
        const float* __restrict__ arow = qk + (size_t)(b * 16 + ml) * FEATD;
        const float* __restrict__ brow = bw + ((size_t)b * NBASE + nl) * FEATD;
        for (int kk = 0; kk < FEATD; kk += 4) {
            v2f a, bb;
            a[0] = arow[kk + kb];  a[1] = arow[kk + kb + 1];
            bb[0] = brow[kk + kb]; bb[1] = brow[kk + kb + 1];
            acc = wmma_f32(a, bb, acc);
        }
    }
    {
        const float* __restrict__ arow = qs + (size_t)(b * 16 + ml) * SEMD;
        const float* __restrict__ brow = bsc + ((size_t)b * NBASE + nl) * SEMD;
        for (int kk = 0; kk < SEMD; kk += 4) {
            v2f a, bb;
            a[0] = arow[kk + kb];  a[1] = arow[kk + kb + 1];
            bb[0] = brow[kk + kb]; bb[1] = brow[kk + kb + 1];
            acc = wmma_f32(a, bb, acc);
        }
    }
    const int mrow = b * 16 + 8 * (lane >> 4);
#pragma unroll
    for (int r = 0; r < 8; ++r)
        score[(size_t)(mrow + r) * NBASE + nl] = acc[r] * INV_TEMP;
}

// softmax over n=8000 for each of 128 rows; grid 128, block 256
__global__ void softmax_kernel(const float* __restrict__ score, float* __restrict__ attn) {
    __shared__ float red[256];
    const int r = blockIdx.x;
    const float* x = score + (size_t)r * NBASE;
    float* y = attn + (size_t)r * NBASE;
    float m = -3.402823466e38f;
    for (int i = threadIdx.x; i < NBASE; i += 256) m = fmaxf(m, x[i]);
    red[threadIdx.x] = m; __syncthreads();
    for (int s = 128; s > 0; s >>= 1) {
        if ((int)threadIdx.x < s) red[threadIdx.x] = fmaxf(red[threadIdx.x], red[threadIdx.x + s]);
        __syncthreads();
    }
    m = red[0]; __syncthreads();
    float sum = 0.0f;
    for (int i = threadIdx.x; i < NBASE; i += 256) {
        float e = expf(x[i] - m);
        y[i] = e;
        sum += e;
    }
    red[threadIdx.x] = sum; __syncthreads();
    for (int s = 128; s > 0; s >>= 1) {
        if ((int)threadIdx.x < s) red[threadIdx.x] += red[threadIdx.x + s];
        __syncthreads();
    }
    const float inv = 1.0f / red[0];
    for (int i = threadIdx.x; i < NBASE; i += 256) y[i] *= inv;
}

// ctx[b,w,d] = attn[b,w,:] @ bw[b,:,d]
// grid (32, 8), block 256 (8 waves). K=8000 split over 8 waves (1000 each),
// 2 interleaved accumulators per wave to break WMMA dependency chains,
// deterministic LDS reduction across waves.
__global__ void ctx_kernel(const float* __restrict__ attn, const float* __restrict__ bw,
                           float* __restrict__ ctx) {
    __shared__ float lds[8 * 256];
    const int lane = threadIdx.x & 31;
    const int wave = threadIdx.x >> 5;
    const int b = blockIdx.y;
    const int n0 = blockIdx.x * 16;
    const int ml = lane & 15;
    const int kb = (lane >> 4) * 2;
    const int nl = n0 + ml;
    v8f acc0 = {}, acc1 = {};
    const float* __restrict__ arow = attn + (size_t)(b * 16 + ml) * NBASE;
    const float* __restrict__ bbase = bw + (size_t)b * NBASE * FEATD;
    const int k0 = wave * 1000;
    for (int kk = k0; kk < k0 + 1000; kk += 8) {
        v2f a, bb;
        a[0] = arow[kk + kb];  a[1] = arow[kk + kb + 1];
        bb[0] = bbase[(size_t)(kk + kb) * FEATD + nl];
        bb[1] = bbase[(size_t)(kk + kb + 1) * FEATD + nl];
        acc0 = wmma_f32(a, bb, acc0);
        a[0] = arow[kk + 4 + kb];  a[1] = arow[kk + 4 + kb + 1];
        bb[0] = bbase[(size_t)(kk + 4 + kb) * FEATD + nl];
        bb[1] = bbase[(size_t)(kk + 4 + kb + 1) * FEATD + nl];
        acc1 = wmma_f32(a, bb, acc1);
    }
#pragma unroll
    for (int r = 0; r < 8; ++r) lds[wave * 256 + r * 32 + lane] = acc0[r] + acc1[r];
    __syncthreads();
    const int e = threadIdx.x;              // 0..255 : one output tile element
    float s = 0.0f;
#pragma unroll
    for (int w = 0; w < 8; ++w) s += lds[w * 256 + e];
    const int r = e >> 5;
    const int le = e & 31;
    const int mrow = b * 16 + 8 * (le >> 4) + r;
    const int col = n0 + (le & 15);
    ctx[(size_t)mrow * FEATD + col] = s;
}

// Wf = w_vs @ fc_w  (512x512x512)   grid (32, 4), block 256 (8 waves)
__global__ void wfuse_kernel(const float* __restrict__ wv, const float* __restrict__ fc,
                             float* __restrict__ wf) {
    const int lane = threadIdx.x & 31;
    const int wave = threadIdx.x >> 5;
    const int m0 = blockIdx.x * 16;
    const int n0 = (blockIdx.y * 8 + wave) * 16;
    const int ml = lane & 15;
    const int kb = (lane >> 4) * 2;
    const int nl = n0 + ml;
    v8f acc = {};
    const float* __restrict__ arow = wv + (size_t)(m0 + ml) * FEATD;
    for (int kk = 0; kk < FEATD; kk += 4) {
        v2f a, bb;
        a[0] = arow[kk + kb];  a[1] = arow[kk + kb + 1];
        bb[0] = fc[(size_t)(kk + kb) * FEATD + nl];
        bb[1] = fc[(size_t)(kk + kb + 1) * FEATD + nl];
        acc = wmma_f32(a, bb, acc);
    }
    const int mrow = m0 + 8 * (lane >> 4);
#pragma unroll
    for (int r = 0; r < 8; ++r) wf[(size_t)(mrow + r) * FEATD + nl] = acc[r];
}

// out[r][d] = ctx[r,:] @ Wf[:,d] + sf[r][d]    grid (2,128), block 256
__global__ void out_kernel(const float* __restrict__ ctx, const float* __restrict__ wf,
                           const float* __restrict__ sf, float* __restrict__ out) {
    const int r = blockIdx.y;
    const int d = blockIdx.x * 256 + threadIdx.x;
    const float* c = ctx + (size_t)r * FEATD;
    float s = sf[(size_t)r * FEATD + d];
    for (int e = 0; e < FEATD; ++e) s += c[e] * wf[(size_t)e * FEATD + d];
    out[(size_t)r * FEATD + d] = s;
}

extern "C" void kernel_launch(void* const* d_in, const int* in_sizes, int n_in,
                              void* d_out, int out_size, void* d_ws, size_t ws_size,
                              hipStream_t stream) {
    (void)in_sizes; (void)n_in; (void)out_size; (void)ws_size;
    const float* support_feat  = (const float*)d_in[0];
    const float* base_weights  = (const float*)d_in[1];
    const float* support_seman = (const float*)d_in[2];
    const float* base_seman    = (const float*)d_in[3];
    const float* map_w1        = (const float*)d_in[4];
    const float* map_b1        = (const float*)d_in[5];
    const float* map_w2        = (const float*)d_in[6];
    const float* map_b2        = (const float*)d_in[7];
    const float* visfuse_w     = (const float*)d_in[8];
    const float* visfuse_b     = (const float*)d_in[9];
    const float* semfuse_w     = (const float*)d_in[10];
    const float* semfuse_b     = (const float*)d_in[11];
    const float* w_qs          = (const float*)d_in[12];
    const float* w_ks          = (const float*)d_in[13];
    const float* w_vs          = (const float*)d_in[14];
    const float* w_qs_sem      = (const float*)d_in[15];
    const float* w_ks_sem      = (const float*)d_in[16];
    const float* fc_w          = (const float*)d_in[17];

    float* out_center = (float*)d_out;                 // (8,16,512) = 65536
    float* out_score  = (float*)d_out + 65536;         // (8,16,8000) = 1,024,000

    // Workspace layout (floats). Needs >= 153.6 MB.
    float* ws  = (float*)d_ws;
    float* h1  = ws;                        // 64000*300 (dead after mapsem2; reused below)
    float* bsc = ws + 19200000;             // 64000*300 calibrated base semantics
    // small buffers carved from the h1 region (h1 is dead by the time these are written)
    float* ssh  = h1;                       // 128*300   support map_sem hidden
    float* ssc  = h1 + 40960;               // 128*300   calibrated support semantics
    float* mean = h1 + 81920;               // 8*812
    float* gate = h1 + 90112;               // 8*812
    float* qt   = h1 + 98304;               // 128*512
    float* qkb  = h1 + 163840;              // 128*512
    float* qsb  = h1 + 229376;              // 128*300
    float* ctxb = h1 + 270336;              // 128*512
    float* wf   = h1 + 335872;              // 512*512
    float* attn = h1 + 598016;              // 128*8000
    float* part = h1 + 1622016;             // 16*8*812 mean partials

    // 1-2: base semantic calibration (WMMA GEMMs, K=300)
    mapsem_kernel<<<dim3(4000, 3), 256, 0, stream>>>(base_seman, map_w1, map_b1, h1, 1);
    mapsem_kernel<<<dim3(4000, 3), 256, 0, stream>>>(h1, map_w2, map_b2, bsc, 0);
    // 3-4: support semantic calibration (same kernel, M=128) — h1 region now reusable
    mapsem_kernel<<<dim3(8, 3), 256, 0, stream>>>(support_seman, map_w1, map_b1, ssh, 1);
    mapsem_kernel<<<dim3(8, 3), 256, 0, stream>>>(ssh, map_w2, map_b2, ssc, 0);
    // 5-6: task memory mean (two-stage deterministic reduction) + fusion gates
    means_partial_kernel<<<dim3(4, 8, 16), 256, 0, stream>>>(base_weights, bsc, part);
    means_final_kernel<<<dim3(4, 8), 256, 0, stream>>>(part, mean);
    gates_kernel<<<dim3(4, 8), 256, 0, stream>>>(mean, visfuse_w, visfuse_b,
                                                 semfuse_w, semfuse_b, gate);
    // 7-8: fused query path: q_tot, then gated key-side transforms pulled onto q
    qtot_kernel<<<dim3(2, 128), 256, 0, stream>>>(support_feat, ssc, w_qs, w_qs_sem, qt);
    qk_kernel<<<dim3(4, 128), 256, 0, stream>>>(qt, gate, w_ks, w_ks_sem, qkb, qsb);
    // 9: attention scores (WMMA, K=512 + K=300) -> attn_score output
    score_kernel<<<dim3(63, 8), 256, 0, stream>>>(qkb, qsb, base_weights, bsc, out_score);
    // 10: softmax over base classes
    softmax_kernel<<<128, 256, 0, stream>>>(out_score, attn);
    // 11: ctx = attn @ base_weights (WMMA, K split over 8 waves + LDS reduce)
    ctx_kernel<<<dim3(32, 8), 256, 0, stream>>>(attn, base_weights, ctxb);
    // 12: fused value/output projection Wf = w_vs @ fc_w (WMMA)
    wfuse_kernel<<<dim3(32, 4), 256, 0, stream>>>(w_vs, fc_w, wf);
    // 13: out = ctx @ Wf + support_feat
    out_kernel<<<dim3(2, 128), 256, 0, stream>>>(ctxb, wf, support_feat, out_center);
}